// GCN_84868553769566
// MI455X (gfx1250) — compile-verified
//
#include <hip/hip_runtime.h>

typedef float v2f __attribute__((ext_vector_type(2)));
typedef float v8f __attribute__((ext_vector_type(8)));

#define F 128  // F_IN == HID == OUT == 128

// ---------------------------------------------------------------------------
// Degree: deg[c] += w over edges, +1.0 self loop per node (PyG gcn_norm).
// ---------------------------------------------------------------------------
__global__ void deg_kernel(const int* __restrict__ col, const float* __restrict__ w,
                           float* __restrict__ deg, int E, int N) {
  int i = blockIdx.x * blockDim.x + threadIdx.x;
  if (i < E) {
    atomicAdd(&deg[col[i]], w[i]);
  } else if (i < E + N) {
    atomicAdd(&deg[i - E], 1.0f);
  }
}

__global__ void dinv_kernel(float* __restrict__ deg, int N) {
  int i = blockIdx.x * blockDim.x + threadIdx.x;
  if (i < N) {
    float d = deg[i];
    deg[i] = d > 0.0f ? rsqrtf(d) : 0.0f;
  }
}

__global__ void norm_kernel(const int* __restrict__ row, const int* __restrict__ col,
                            const float* __restrict__ w, const float* __restrict__ dinv,
                            float* __restrict__ normE, float* __restrict__ normSelf,
                            int E, int N) {
  int i = blockIdx.x * blockDim.x + threadIdx.x;
  if (i < E) {
    normE[i] = dinv[row[i]] * w[i] * dinv[col[i]];
  } else if (i < E + N) {
    float d = dinv[i - E];
    normSelf[i - E] = d * d;  // self-loop weight is 1.0
  }
}

// ---------------------------------------------------------------------------
// Y[N x 128] = X[N x 128] @ W[128 x 128] using V_WMMA_F32_16X16X4_F32.
// 256 threads = 8 waves; each wave owns a 16-row strip and all 128 columns
// (8 accumulator tiles of 16x16 f32 = 64 VGPRs).
//
// W is staged in LDS K-INTERLEAVED: element W[k][n] lives at
//   sW[(k>>1)*256 + 2*n + (k&1)]
// so a B fragment {W[k][n], W[k+1][n]} is one aligned 8-byte ds_load_b64
// that lands directly in an even VGPR pair (no shuffle movs).
//
// Fragment layouts per CDNA5 ISA 7.12.2 (f32, wave32):
//   A 16x4 : lanes 0-15 hold M=lane, v0=K0 v1=K1; lanes 16-31 hold M=lane-16,
//            v0=K2 v1=K3.
//   B 4x16 : lanes 0-15 hold N=lane, v0=K0 v1=K1; lanes 16-31 hold N=lane-16,
//            v0=K2 v1=K3.
//   C/D    : lane<16 -> vj = D[M=j][N=lane]; lane>=16 -> vj = D[M=8+j][N=lane-16].
//
// Out-of-range rows (last block) load a clamped valid row instead of being
// predicated off (WMMA needs EXEC all-ones anyway); their D rows are simply
// not stored, so no exec-mask toggling in the hot loop.
// ---------------------------------------------------------------------------
__global__ __launch_bounds__(256) void gemm_wmma(const float* __restrict__ X,
                                                 const float* __restrict__ W,
                                                 float* __restrict__ Y, int N) {
  __shared__ float sW[F * F];  // 64 KB, k-interleaved pair layout
  {
    const float4* w4 = (const float4*)W;
    int tid = threadIdx.x;
#pragma unroll
    for (int i = 0; i < (F * F / 4) / 256; ++i) {
      int idx = tid + i * 256;       // float4 index into row-major W
      int kk = idx >> 5;             // row k  (128 floats = 32 float4 per row)
      int nn = (idx & 31) << 2;      // starting column
      float4 v = w4[idx];
      float* dst = &sW[(kk >> 1) * 256 + (kk & 1) + nn * 2];
      dst[0] = v.x;
      dst[2] = v.y;
      dst[4] = v.z;
      dst[6] = v.w;
    }
  }
  __syncthreads();

  const int wave  = threadIdx.x >> 5;
  const int lane  = threadIdx.x & 31;
  const int r0    = (blockIdx.x * 8 + wave) * 16;
  const int m     = lane & 15;
  const int khalf = lane >> 4;        // 0: K{0,1}  1: K{2,3}
  const int koff  = khalf << 1;

  int arow = r0 + m;
  if (arow >= N) arow = N - 1;        // clamp: garbage rows never stored
  const float* xrow = X + (size_t)arow * F + koff;

  v8f acc[8] = {};

#pragma unroll 4
  for (int k = 0; k < F; k += 4) {
    v2f a = *(const v2f*)(xrow + k);                       // global b64
    const float* wb = &sW[((k >> 1) + khalf) * 256 + m * 2];
#pragma unroll
    for (int t = 0; t < 8; ++t) {
      v2f b = *(const v2f*)(wb + 32 * t);                  // ds_load_b64
      acc[t] = __builtin_amdgcn_wmma_f32_16x16x4_f32(
          /*neg_a=*/false, a, /*neg_b=*/false, b,
          /*c_mod=*/(short)0, acc[t], /*reuse_a=*/false, /*reuse_b=*/false);
    }
  }

  const int rAdd = khalf * 8;
#pragma unroll
  for (int t = 0; t < 8; ++t) {
#pragma unroll
    for (int j = 0; j < 8; ++j) {
      int r = r0 + rAdd + j;
      if (r < N) Y[(size_t)r * F + 16 * t + m] = acc[t][j];
    }
  }
}

// ---------------------------------------------------------------------------
// Scatter-add: one wave per (edge | self loop). float4 gather of the 512 B
// source row (L2-resident: 51.2 MB << 192 MB L2), scale, 4 float atomics/lane.
// ---------------------------------------------------------------------------
__global__ void scatter_kernel(const float* __restrict__ H,
                               const int* __restrict__ row, const int* __restrict__ col,
                               const float* __restrict__ normE,
                               const float* __restrict__ normSelf,
                               float* __restrict__ out, int E, int N) {
  int gid  = blockIdx.x * blockDim.x + threadIdx.x;
  int wave = gid >> 5;
  int lane = gid & 31;
  int r, c; float s;
  if (wave < E) {
    r = row[wave]; c = col[wave]; s = normE[wave];
  } else if (wave < E + N) {
    r = c = wave - E; s = normSelf[r];
  } else {
    return;
  }
  float4 v = ((const float4*)(H + (size_t)r * F))[lane];
  float* dst = out + (size_t)c * F + lane * 4;
  atomicAdd(dst + 0, s * v.x);
  atomicAdd(dst + 1, s * v.y);
  atomicAdd(dst + 2, s * v.z);
  atomicAdd(dst + 3, s * v.w);
}

// ---------------------------------------------------------------------------
// Vectorized epilogues: buf is [N x 128] row-major, bias is 128 = 32 float4.
// ---------------------------------------------------------------------------
__global__ void bias_relu_kernel(float4* __restrict__ buf, const float4* __restrict__ b,
                                 size_t total4) {
  size_t i = (size_t)blockIdx.x * blockDim.x + threadIdx.x;
  if (i < total4) {
    float4 v = buf[i];
    float4 bb = b[i & 31];
    v.x = fmaxf(v.x + bb.x, 0.0f);
    v.y = fmaxf(v.y + bb.y, 0.0f);
    v.z = fmaxf(v.z + bb.z, 0.0f);
    v.w = fmaxf(v.w + bb.w, 0.0f);
    buf[i] = v;
  }
}

__global__ void bias_kernel(float4* __restrict__ buf, const float4* __restrict__ b,
                            size_t total4) {
  size_t i = (size_t)blockIdx.x * blockDim.x + threadIdx.x;
  if (i < total4) {
    float4 v = buf[i];
    float4 bb = b[i & 31];
    v.x += bb.x; v.y += bb.y; v.z += bb.z; v.w += bb.w;
    buf[i] = v;
  }
}

// ---------------------------------------------------------------------------
extern "C" void kernel_launch(void* const* d_in, const int* in_sizes, int n_in,
                              void* d_out, int out_size, void* d_ws, size_t ws_size,
                              hipStream_t stream) {
  const float* x  = (const float*)d_in[0];
  const int*   ei = (const int*)d_in[1];
  const float* ew = (const float*)d_in[2];
  const float* W1 = (const float*)d_in[3];
  const float* b1 = (const float*)d_in[4];
  const float* W2 = (const float*)d_in[5];
  const float* b2 = (const float*)d_in[6];

  const int N = in_sizes[0] / F;
  const int E = in_sizes[2];
  const int* rowIdx = ei;       // edge_index[0,:]
  const int* colIdx = ei + E;   // edge_index[1,:]

  // workspace carve-up (16B-aligned float offsets)
  auto up4 = [](size_t v) { return (v + 3) & ~(size_t)3; };
  float* deg      = (float*)d_ws;
  float* normE    = deg + up4((size_t)N);
  float* normSelf = normE + up4((size_t)E);
  float* bufA     = normSelf + up4((size_t)N);
  float* bufB     = bufA + (size_t)N * F;
  float* out      = (float*)d_out;

  const size_t featBytes = (size_t)N * F * sizeof(float);
  hipMemsetAsync(deg, 0, (size_t)N * sizeof(float), stream);
  hipMemsetAsync(bufB, 0, featBytes, stream);
  hipMemsetAsync(d_out, 0, featBytes, stream);

  const int tb = 256;
  const int EN = E + N;
  deg_kernel<<<(EN + tb - 1) / tb, tb, 0, stream>>>(colIdx, ew, deg, E, N);
  dinv_kernel<<<(N + tb - 1) / tb, tb, 0, stream>>>(deg, N);
  norm_kernel<<<(EN + tb - 1) / tb, tb, 0, stream>>>(rowIdx, colIdx, ew, deg,
                                                     normE, normSelf, E, N);

  const int gBlocks = (N + 127) / 128;  // 128 rows per block (8 waves x 16)
  const long long sThreads = (long long)EN * 32;
  const int sBlocks = (int)((sThreads + tb - 1) / tb);
  const size_t total4 = (size_t)N * (F / 4);
  const int eBlocks = (int)((total4 + tb - 1) / tb);

  // Layer 1: h1 = x @ W1 ; agg1 = scatter(norm * h1[row] -> col) ; relu(+b1)
  gemm_wmma<<<gBlocks, 256, 0, stream>>>(x, W1, bufA, N);
  scatter_kernel<<<sBlocks, tb, 0, stream>>>(bufA, rowIdx, colIdx, normE, normSelf,
                                             bufB, E, N);
  bias_relu_kernel<<<eBlocks, tb, 0, stream>>>((float4*)bufB, (const float4*)b1,
                                               total4);

  // Layer 2: h2 = agg1 @ W2 ; out = scatter(...) + b2
  gemm_wmma<<<gBlocks, 256, 0, stream>>>(bufB, W2, bufA, N);
  scatter_kernel<<<sBlocks, tb, 0, stream>>>(bufA, rowIdx, colIdx, normE, normSelf,
                                             out, E, N);
  bias_kernel<<<eBlocks, tb, 0, stream>>>((float4*)out, (const float4*)b2, total4);
}